// FreqMoE_38929583571295
// MI455X (gfx1250) — compile-verified
//
#include <hip/hip_runtime.h>
#include <hip/hip_bf16.h>

#define NFFT   4096
#define LOG2N  12
#define FREQ   2049
#define BB     32
#define CC     8
#define EE     8

typedef float v2f __attribute__((ext_vector_type(2)));
typedef float v8f __attribute__((ext_vector_type(8)));

// ---------------------------------------------------------------------------
// In-place radix-2 DIT FFT over a bit-reversed-filled LDS buffer.
// sign = -1 forward, +1 inverse (caller scales by 1/N for inverse).
// 256 threads, 2048 butterflies per stage (8 per thread).
// ---------------------------------------------------------------------------
__device__ inline void fft_inplace(float2* buf, int tid, float sign) {
  for (int st = 1; st <= LOG2N; ++st) {
    const int half = 1 << (st - 1);
    const float ang_step = sign * 6.28318530717958647692f / (float)(1 << st);
    for (int t = tid; t < NFFT / 2; t += 256) {
      const int j  = t & (half - 1);
      const int g  = t >> (st - 1);
      const int i1 = (g << st) + j;
      const int i2 = i1 + half;
      float sn, cs;
      __sincosf(ang_step * (float)j, &sn, &cs);
      const float2 u = buf[i1];
      const float2 v = buf[i2];
      const float vr = v.x * cs - v.y * sn;
      const float vi = v.x * sn + v.y * cs;
      buf[i1] = make_float2(u.x + vr, u.y + vi);
      buf[i2] = make_float2(u.x - vr, u.y - vi);
    }
    __syncthreads();
  }
}

// ---------------------------------------------------------------------------
// K0: boundaries + integer band edges (tiny, single thread).
// ---------------------------------------------------------------------------
__global__ void k_bounds(const float* __restrict__ raw,
                         float* __restrict__ out_bounds,
                         int* __restrict__ idx) {
  if (threadIdx.x != 0 || blockIdx.x != 0) return;
  float s[7];
  for (int i = 0; i < 7; ++i) s[i] = 1.f / (1.f + __expf(-raw[i]));
  for (int i = 1; i < 7; ++i) {           // insertion sort ascending
    float key = s[i];
    int j = i - 1;
    while (j >= 0 && s[j] > key) { s[j + 1] = s[j]; --j; }
    s[j + 1] = key;
  }
  float b[9];
  b[0] = 0.f;
  for (int i = 0; i < 7; ++i) b[i + 1] = s[i];
  b[8] = 1.f;
  for (int i = 0; i < 9; ++i) out_bounds[i] = b[i];
  for (int i = 0; i < 8; ++i) idx[i] = (int)floorf(b[i] * (float)FREQ);
  idx[8] = FREQ;
}

// ---------------------------------------------------------------------------
// K1: per-(b,c) instance norm + forward rfft. One workgroup per signal.
// ---------------------------------------------------------------------------
__global__ __launch_bounds__(256) void k_norm_rfft(const float* __restrict__ x,
                                                   float* __restrict__ freq,
                                                   float* __restrict__ stats) {
  __shared__ float2 buf[NFFT];    // 32 KB
  __shared__ float red[256];
  const int bc = blockIdx.x;
  const int tid = threadIdx.x;
  const float* xp = x + (size_t)bc * NFFT;

  float vloc[16];
  float sum = 0.f, sq = 0.f;
#pragma unroll
  for (int k = 0; k < 16; ++k) {
    if (k < 12) __builtin_prefetch(&xp[tid + 256 * (k + 4)], 0, 1);
    const float v = xp[tid + 256 * k];
    vloc[k] = v; sum += v; sq += v * v;
  }
  red[tid] = sum; __syncthreads();
  for (int s = 128; s > 0; s >>= 1) { if (tid < s) red[tid] += red[tid + s]; __syncthreads(); }
  const float mean = red[0] / (float)NFFT;
  __syncthreads();
  red[tid] = sq; __syncthreads();
  for (int s = 128; s > 0; s >>= 1) { if (tid < s) red[tid] += red[tid + s]; __syncthreads(); }
  const float var = (red[0] - (float)NFFT * mean * mean) / (float)(NFFT - 1) + 1e-5f;
  const float rstd = rsqrtf(var);
  if (tid == 0) { stats[bc * 2 + 0] = mean; stats[bc * 2 + 1] = sqrtf(var); }

#pragma unroll
  for (int k = 0; k < 16; ++k) {
    const int i = tid + 256 * k;
    const int r = (int)(__brev((unsigned)i) >> (32 - LOG2N));
    buf[r] = make_float2((vloc[k] - mean) * rstd, 0.f);
  }
  __syncthreads();
  fft_inplace(buf, tid, -1.f);
  for (int f = tid; f < FREQ; f += 256) {
    freq[((size_t)bc * FREQ + f) * 2 + 0] = buf[f].x;
    freq[((size_t)bc * FREQ + f) * 2 + 1] = buf[f].y;
  }
}

// ---------------------------------------------------------------------------
// K2: gating input = mean over channels of |freq|.
// ---------------------------------------------------------------------------
__global__ __launch_bounds__(256) void k_gate_in(const float* __restrict__ freq,
                                                 float* __restrict__ gin) {
  const int i = blockIdx.x * 256 + threadIdx.x;
  if (i >= BB * FREQ) return;
  const int b = i / FREQ, f = i % FREQ;
  float acc = 0.f;
#pragma unroll
  for (int c = 0; c < CC; ++c) {
    const size_t o = (((size_t)(b * CC + c)) * FREQ + f) * 2;
    const float re = freq[o], im = freq[o + 1];
    acc += sqrtf(re * re + im * im);
  }
  gin[(size_t)b * FREQ + f] = acc * (1.f / (float)CC);
}

// ---------------------------------------------------------------------------
// K3: H = relu(G @ W1^T + b1) via V_WMMA_F32_16X16X4_F32.
// One 16x16 output tile per wave. Main K-loop (512 iters) has NO bounds
// checks: k+1 <= 2047 < 2049 always, and out-of-range N columns are handled
// by clamping the B row pointer (their accumulator columns are never stored,
// and D column n depends only on B column n). Single masked tail for k=2048
// uses unconditional loads + cndmask, so no predicated memory ops anywhere.
// A 16x4 layout: lanes 0-15 -> {K0,K1}, lanes 16-31 -> {K2,K3} (M = lane%16).
// B 4x16 layout: VGPR0/1 -> rows {K0,K1} (lanes 0-15) / {K2,K3} (lanes 16-31),
// N = lane%16.  B[k][n] = W1[n][k].
// ---------------------------------------------------------------------------
__global__ __launch_bounds__(256) void k_mlp1_wmma(const float* __restrict__ G,
                                                   const float* __restrict__ W1,
                                                   const float* __restrict__ b1,
                                                   float* __restrict__ H) {
  const int wave = (int)((blockIdx.x * blockDim.x + threadIdx.x) >> 5);
  const int lane = threadIdx.x & 31;
  const int NT = (FREQ + 15) / 16;              // 129 tiles in N
  if (wave >= 2 * NT) return;                   // wave-uniform: EXEC all-1s below
  const int mt = wave / NT;
  const int nt = wave % NT;
  const int m0 = mt * 16, n0 = nt * 16;
  const int half = lane >> 4;
  const int l = lane & 15;
  const int m = m0 + l;                         // < 32 always
  const int n = n0 + l;
  const int nsafe = (n < FREQ) ? n : (FREQ - 1);

  const float* Ar = G  + (size_t)m * FREQ + 2 * half;
  const float* Br = W1 + (size_t)nsafe * FREQ + 2 * half;

  v8f acc = {0.f, 0.f, 0.f, 0.f, 0.f, 0.f, 0.f, 0.f};
#pragma unroll 4
  for (int k0 = 0; k0 < 2048; k0 += 4) {
    v2f a;  a.x = Ar[k0];  a.y = Ar[k0 + 1];    // one global_load_b64
    v2f bm; bm.x = Br[k0]; bm.y = Br[k0 + 1];   // one global_load_b64
    acc = __builtin_amdgcn_wmma_f32_16x16x4_f32(false, a, false, bm,
                                                (short)0, acc, false, false);
  }
  { // K tail: only k = 2048 exists; lanes 16-31 (K2,K3) contribute zero.
    const float ga = G[(size_t)m * FREQ + (FREQ - 1)];
    const float wa = W1[(size_t)nsafe * FREQ + (FREQ - 1)];
    v2f a;  a.x = half ? 0.f : ga;  a.y = 0.f;
    v2f bm; bm.x = half ? 0.f : wa; bm.y = 0.f;
    acc = __builtin_amdgcn_wmma_f32_16x16x4_f32(false, a, false, bm,
                                                (short)0, acc, false, false);
  }
  // D layout: VGPR r -> M = r (lanes 0-15) / 8+r (lanes 16-31), N = lane%16.
  if (n < FREQ) {
    const float bias = b1[n];
#pragma unroll
    for (int r = 0; r < 8; ++r) {
      const int mm = m0 + r + 8 * half;
      const float v = acc[r] + bias;
      H[(size_t)mm * FREQ + n] = v > 0.f ? v : 0.f;
    }
  }
}

// ---------------------------------------------------------------------------
// K4: logits = H @ W2^T + b2, then softmax. One block per batch row.
// ---------------------------------------------------------------------------
__global__ __launch_bounds__(256) void k_gate_scores(const float* __restrict__ H,
                                                     const float* __restrict__ W2,
                                                     const float* __restrict__ b2,
                                                     float* __restrict__ scores) {
  __shared__ float red[256];
  __shared__ float logits[8];
  const int b = blockIdx.x, tid = threadIdx.x;
  float acc[8] = {0.f, 0.f, 0.f, 0.f, 0.f, 0.f, 0.f, 0.f};
  for (int f = tid; f < FREQ; f += 256) {
    const float hv = H[(size_t)b * FREQ + f];
#pragma unroll
    for (int e = 0; e < 8; ++e) acc[e] += hv * W2[(size_t)e * FREQ + f];
  }
  for (int e = 0; e < 8; ++e) {
    red[tid] = acc[e]; __syncthreads();
    for (int s = 128; s > 0; s >>= 1) { if (tid < s) red[tid] += red[tid + s]; __syncthreads(); }
    if (tid == 0) logits[e] = red[0] + b2[e];
    __syncthreads();
  }
  if (tid == 0) {
    float m = logits[0];
    for (int e = 1; e < 8; ++e) m = fmaxf(m, logits[e]);
    float sum = 0.f, ex[8];
    for (int e = 0; e < 8; ++e) { ex[e] = __expf(logits[e] - m); sum += ex[e]; }
    for (int e = 0; e < 8; ++e) scores[b * 8 + e] = ex[e] / sum;
  }
}

// ---------------------------------------------------------------------------
// K5: inverse transforms. Blocks [0, B*E*C): expert irfft (band mask).
// Blocks [B*E*C, B*E*C + B*C): gated combined irfft + denormalize.
// ---------------------------------------------------------------------------
__global__ __launch_bounds__(256) void k_irfft(const float* __restrict__ freq,
                                               const int* __restrict__ idx,
                                               const float* __restrict__ gates,
                                               const float* __restrict__ stats,
                                               float* __restrict__ out_comb,
                                               float* __restrict__ out_exp) {
  __shared__ float2 buf[NFFT];    // 32 KB
  __shared__ int sIdx[9];
  __shared__ float sg[8];
  const int id = blockIdx.x;
  const int tid = threadIdx.x;
  const bool comb = (id >= BB * EE * CC);
  int b, c, e = 0;
  if (!comb) { b = id >> 6; e = (id >> 3) & 7; c = id & 7; }
  else       { const int q = id - BB * EE * CC; b = q >> 3; c = q & 7; }
  if (tid < 9) sIdx[tid] = idx[tid];
  if (tid < 8) sg[tid] = gates[b * 8 + tid];
  __syncthreads();
  const int lo = comb ? 0 : sIdx[e];
  const int hi = comb ? 0 : sIdx[e + 1];

  const float* fp = freq + ((size_t)(b * CC + c) * FREQ) * 2;
  for (int f = tid; f < FREQ; f += 256) {
    __builtin_prefetch(&fp[2 * (f + 256)], 0, 1);
    float re = fp[2 * f], im = fp[2 * f + 1];
    float w;
    if (!comb) {
      w = (f >= lo && f < hi) ? 1.f : 0.f;
    } else {
      int band = 0;
#pragma unroll
      for (int j = 1; j < 8; ++j) band += (f >= sIdx[j]) ? 1 : 0;
      w = sg[band];
    }
    re *= w; im *= w;
    if (f == 0 || f == NFFT / 2) im = 0.f;
    const int r1 = (int)(__brev((unsigned)f) >> (32 - LOG2N));
    buf[r1] = make_float2(re, im);
    if (f >= 1 && f < NFFT / 2) {
      const int r2 = (int)(__brev((unsigned)(NFFT - f)) >> (32 - LOG2N));
      buf[r2] = make_float2(re, -im);    // Hermitian mirror
    }
  }
  __syncthreads();
  fft_inplace(buf, tid, +1.f);
  const float scale = 1.f / (float)NFFT;
  if (!comb) {
    float* op = out_exp + (size_t)id * NFFT;   // id == ((b*E+e)*C + c)
    for (int t = tid; t < NFFT; t += 256) op[t] = buf[t].x * scale;
  } else {
    const int bc = b * CC + c;
    const float mean = stats[bc * 2 + 0];
    const float sqv  = stats[bc * 2 + 1];
    float* op = out_comb + (size_t)bc * NFFT;
    for (int t = tid; t < NFFT; t += 256) op[t] = buf[t].x * scale * sqv + mean;
  }
}

// ---------------------------------------------------------------------------
extern "C" void kernel_launch(void* const* d_in, const int* in_sizes, int n_in,
                              void* d_out, int out_size, void* d_ws, size_t ws_size,
                              hipStream_t stream) {
  (void)in_sizes; (void)n_in; (void)out_size; (void)ws_size;
  const float* x   = (const float*)d_in[0];
  const float* raw = (const float*)d_in[1];
  const float* W1  = (const float*)d_in[2];
  const float* b1  = (const float*)d_in[3];
  const float* W2  = (const float*)d_in[4];
  const float* b2  = (const float*)d_in[5];

  float* out = (float*)d_out;
  float* out_comb   = out;                 // [32,8,4096]  = 1048576
  float* out_bounds = out + 1048576;       // [9]
  float* out_scores = out + 1048585;       // [32,8]       = 256
  float* out_exp    = out + 1048841;       // [32,8,8,4096]= 8388608

  float* ws    = (float*)d_ws;
  float* freq  = ws;                                   // B*C*F*2 = 1049088 fl
  float* stats = freq + (size_t)BB * CC * FREQ * 2;    // 512 fl
  float* gin   = stats + 512;                          // B*F = 65568 fl
  float* H     = gin + (size_t)BB * FREQ;              // B*F = 65568 fl
  int*   idx   = (int*)(H + (size_t)BB * FREQ);        // 9 ints

  k_bounds<<<1, 32, 0, stream>>>(raw, out_bounds, idx);
  k_norm_rfft<<<BB * CC, 256, 0, stream>>>(x, freq, stats);
  k_gate_in<<<(BB * FREQ + 255) / 256, 256, 0, stream>>>(freq, gin);
  {
    const int total_waves = 2 * ((FREQ + 15) / 16);    // 258
    const int blocks = (total_waves * 32 + 255) / 256; // 33
    k_mlp1_wmma<<<blocks, 256, 0, stream>>>(gin, W1, b1, H);
  }
  k_gate_scores<<<BB, 256, 0, stream>>>(H, W2, b2, out_scores);
  k_irfft<<<BB * EE * CC + BB * CC, 256, 0, stream>>>(freq, idx, out_scores,
                                                      stats, out_comb, out_exp);
}